// Sparsemax_13580686590267
// MI455X (gfx1250) — compile-verified
//
#include <hip/hip_runtime.h>
#include <stdint.h>

// Row length fixed by the reference: x is (4096, 8192) fp32, sparsemax over axis -1.
#define D_LEN 8192
#define TPB 256
#define EPT (D_LEN / TPB)       // 32 elements per thread
#define V4N (EPT / 4)           // 8 float4 per thread
#define ROWS_PER_BLOCK 4
#define NWAVES (TPB / 32)       // 8 waves (wave32)

typedef unsigned int u32x4 __attribute__((ext_vector_type(4)));
typedef int          s32x8 __attribute__((ext_vector_type(8)));
typedef int          s32x4 __attribute__((ext_vector_type(4)));

#define HAS_TDM __has_builtin(__builtin_amdgcn_tensor_load_to_lds)

#if HAS_TDM
// Issue a TDM DMA of one row (D_LEN fp32, contiguous) from global memory into LDS.
// D# layout per CDNA5 ISA:
//   group0: [1:0]=count(1), [63:32]=lds_addr, [120:64]=global_addr, [127:126]=type(2)
//   group1: [17:16]=data_size(2 -> 4B), [79:48]=tensor_dim0, [111:80]=tensor_dim1,
//           [127:112]=tile_dim0, [143:128]=tile_dim1, [207:160]=tensor_dim0_stride
// This toolchain (clang-23 / therock-10.0 headers) uses the 6-arg builtin:
//   (u32x4 g0, s32x8 g1, s32x4 g2, s32x4 g3, s32x8 g4, i32 cpol)
__device__ __forceinline__ void tdm_load_row(const float* gsrc, uint32_t lds_byte_off) {
  uint64_t ga = (uint64_t)(uintptr_t)gsrc;
  u32x4 g0;
  g0[0] = 1u;                                                   // count=1, user mode
  g0[1] = lds_byte_off;                                         // lds_addr (bytes)
  g0[2] = (uint32_t)ga;                                         // global_addr[31:0]
  g0[3] = (uint32_t)((ga >> 32) & 0x01FFFFFFu) | 0x80000000u;   // addr[56:32] | type=2

  s32x8 g1;
  g1[0] = 0x00020000;                                    // wg_mask=0, data_size=2 (4B)
  g1[1] = (int)(((uint32_t)D_LEN & 0xFFFFu) << 16);      // tensor_dim0[15:0] in [31:16]
  g1[2] = (int)(((uint32_t)D_LEN >> 16) | (1u << 16));   // tensor_dim0[31:16] | tensor_dim1[15:0]=1
  g1[3] = (int)(((uint32_t)D_LEN & 0xFFFFu) << 16);      // tensor_dim1[31:16]=0 | tile_dim0
  g1[4] = 1;                                             // tile_dim1=1, tile_dim2=0
  g1[5] = D_LEN;                                         // tensor_dim0_stride[31:0]
  g1[6] = 0;                                             // stride0[47:32] | stride1[15:0]
  g1[7] = 0;

  s32x4 z4 = {0, 0, 0, 0};                               // groups 2/3 unused (<=2D)
  s32x8 z8 = {0, 0, 0, 0, 0, 0, 0, 0};
  __builtin_amdgcn_tensor_load_to_lds(g0, g1, z4, z4, z8, 0);
}
#endif

__global__ __launch_bounds__(TPB) void sparsemax_kernel(const float* __restrict__ x,
                                                        float* __restrict__ out,
                                                        int nrows) {
  extern __shared__ float dynbuf[];
  float* buf0 = dynbuf;                  // row buffer A (32 KB)
  float* buf1 = dynbuf + D_LEN;          // row buffer B (32 KB)
  __shared__ float red_s[NWAVES];
  __shared__ float red_k[NWAVES];

  const int tid  = threadIdx.x;
  const int wave = tid >> 5;
  const int lane = tid & 31;
  const int row0 = blockIdx.x * ROWS_PER_BLOCK;

#if HAS_TDM
  if (wave == 0 && row0 < nrows) {
    tdm_load_row(x + (size_t)row0 * D_LEN, (uint32_t)(uintptr_t)buf0);
  }
#endif

  for (int r = 0; r < ROWS_PER_BLOCK; ++r) {
    const int row = row0 + r;
    if (row >= nrows) break;
    float* cur = (r & 1) ? buf1 : buf0;
#if HAS_TDM
    float* nxt = (r & 1) ? buf0 : buf1;
    const bool have_next = (r + 1 < ROWS_PER_BLOCK) && (row + 1 < nrows);
    if (wave == 0) {
      if (have_next) {
        // Prefetch next row via TDM while we compute on the current one,
        // then wait until only that new DMA is outstanding => current row done.
        tdm_load_row(x + (size_t)(row + 1) * D_LEN, (uint32_t)(uintptr_t)nxt);
        __builtin_amdgcn_s_wait_tensorcnt(1);
      } else {
        __builtin_amdgcn_s_wait_tensorcnt(0);
      }
    }
#else
    {   // fallback: cooperative coalesced copy global -> LDS
      const float4* g = (const float4*)(x + (size_t)row * D_LEN);
      float4* l = (float4*)cur;
      #pragma unroll
      for (int j = 0; j < V4N; ++j) l[tid + j * TPB] = g[tid + j * TPB];
    }
#endif
    __syncthreads();

    // Pull this thread's 32 elements into registers (float4 LDS reads).
    float v[EPT];
    #pragma unroll
    for (int j = 0; j < V4N; ++j) {
      float4 q = ((const float4*)cur)[tid + j * TPB];
      v[4 * j + 0] = q.x; v[4 * j + 1] = q.y;
      v[4 * j + 2] = q.z; v[4 * j + 3] = q.w;
    }

    // ---- initial tau: (sum - 1) / d  (first Michelot iterate from full support)
    float s = 0.f;
    #pragma unroll
    for (int i = 0; i < EPT; ++i) s += v[i];
    #pragma unroll
    for (int off = 16; off >= 1; off >>= 1) s += __shfl_xor(s, off, 32);
    if (lane == 0) red_s[wave] = s;
    __syncthreads();
    float total = 0.f;
    #pragma unroll
    for (int w = 0; w < NWAVES; ++w) total += red_s[w];
    __syncthreads();

    float tau = (total - 1.0f) / (float)D_LEN;

    // ---- Michelot fixed point: tau <- (sum_{x>tau} x - 1) / #{x>tau}
    // Monotone, converges finitely; break is uniform (all threads see identical S,K).
    for (int it = 0; it < 64; ++it) {
      float ps = 0.f, pk = 0.f;
      #pragma unroll
      for (int i = 0; i < EPT; ++i) {
        if (v[i] > tau) { ps += v[i]; pk += 1.0f; }
      }
      #pragma unroll
      for (int off = 16; off >= 1; off >>= 1) {
        ps += __shfl_xor(ps, off, 32);
        pk += __shfl_xor(pk, off, 32);
      }
      if (lane == 0) { red_s[wave] = ps; red_k[wave] = pk; }
      __syncthreads();
      float S = 0.f, K = 0.f;
      #pragma unroll
      for (int w = 0; w < NWAVES; ++w) { S += red_s[w]; K += red_k[w]; }
      __syncthreads();
      K = fmaxf(K, 1.0f);
      float tnew = (S - 1.0f) / K;
      if (tnew == tau) break;
      tau = tnew;
    }

    // ---- emit relu(x - tau), non-finite -> NaN (matches reference masking)
    float* orow = out + (size_t)row * D_LEN;
    #pragma unroll
    for (int j = 0; j < V4N; ++j) {
      float4 q;
      float t0 = v[4 * j + 0] - tau; t0 = t0 > 0.f ? t0 : 0.f;
      float t1 = v[4 * j + 1] - tau; t1 = t1 > 0.f ? t1 : 0.f;
      float t2 = v[4 * j + 2] - tau; t2 = t2 > 0.f ? t2 : 0.f;
      float t3 = v[4 * j + 3] - tau; t3 = t3 > 0.f ? t3 : 0.f;
      q.x = __builtin_isfinite(t0) ? t0 : __builtin_nanf("");
      q.y = __builtin_isfinite(t1) ? t1 : __builtin_nanf("");
      q.z = __builtin_isfinite(t2) ? t2 : __builtin_nanf("");
      q.w = __builtin_isfinite(t3) ? t3 : __builtin_nanf("");
      ((float4*)orow)[tid + j * TPB] = q;
    }
    __syncthreads();   // all reads of red arrays / buffers done before next round
  }
}

extern "C" void kernel_launch(void* const* d_in, const int* in_sizes, int n_in,
                              void* d_out, int out_size, void* d_ws, size_t ws_size,
                              hipStream_t stream) {
  (void)n_in; (void)out_size; (void)d_ws; (void)ws_size;
  const float* x = (const float*)d_in[0];
  float* out = (float*)d_out;
  const int nrows = in_sizes[0] / D_LEN;
  const int grid = (nrows + ROWS_PER_BLOCK - 1) / ROWS_PER_BLOCK;
  const size_t shmem = (size_t)2 * D_LEN * sizeof(float);   // double-buffered row storage
  sparsemax_kernel<<<grid, TPB, shmem, stream>>>(x, out, nrows);
}